// ChamferIndex_1486058684543
// MI455X (gfx1250) — compile-verified
//
#include <hip/hip_runtime.h>

typedef float v2f __attribute__((ext_vector_type(2)));
typedef float v8f __attribute__((ext_vector_type(8)));

#define NPTS   16384
#define NB     8
#define CH     2048                      // candidate points per LDS chunk
#define NCHUNK (NPTS / CH)               // 8
#define THREADS 256
#define WAVES  (THREADS / 32)            // 8
#define TPW    2                         // query tiles (of 16) per wave
#define TILES_PER_WG (WAVES * TPW)       // 16
#define WG_PER_PAIR  (NPTS / 16 / TILES_PER_WG)  // 64
#define PPT    (CH / THREADS)            // prefetch points per thread = 8
#define IPC    (CH / 16)                 // WMMA iterations per chunk = 128

__global__ __launch_bounds__(THREADS)
void chamfer_nn(const float* __restrict__ xyz1,
                const float* __restrict__ xyz2,
                int* __restrict__ out)
{
    // double-buffered candidate chunk: {-2*y0, -2*y1, -2*y2, ||y||^2}
    __shared__ float4 lbuf[2][CH];       // 64 KB

    const int pair = blockIdx.x / WG_PER_PAIR;   // (batch, direction)
    const int wgp  = blockIdx.x % WG_PER_PAIR;
    const int b    = pair >> 1;
    const int dir  = pair & 1;

    const float* __restrict__ xq = (dir ? xyz2 : xyz1) + (size_t)b * NPTS * 3; // queries
    const float* __restrict__ yc = (dir ? xyz1 : xyz2) + (size_t)b * NPTS * 3; // candidates
    int* __restrict__ outp = out + (size_t)dir * NB * NPTS + (size_t)b * NPTS;

    const int  tid   = threadIdx.x;
    const int  lane  = tid & 31;
    const int  wave  = tid >> 5;
    const int  nlane = lane & 15;        // column n of C/D tile held by this lane
    const bool lo    = (lane < 16);

    // ---- A matrices (16x4, K padded with 0) for this wave's query tiles ----
    v2f av[TPW];
    int qb[TPW];
#pragma unroll
    for (int t = 0; t < TPW; ++t) {
        qb[t] = (wgp * TILES_PER_WG + wave * TPW + t) * 16;
        const float* q = xq + (size_t)(qb[t] + nlane) * 3;
        float q0 = q[0], q1 = q[1], q2 = q[2];
        av[t][0] = lo ? q0 : q2;         // K=0 (lanes 0-15) / K=2 (lanes 16-31)
        av[t][1] = lo ? q1 : 0.0f;       // K=1 / K=3 (zero pad)
    }

    // running best (distance, wmma-iteration) per C/D register row
    float bd[TPW][8];
    int   bi[TPW][8];
#pragma unroll
    for (int t = 0; t < TPW; ++t)
#pragma unroll
        for (int r = 0; r < 8; ++r) { bd[t][r] = 3.4e38f; bi[t][r] = 0; }

    // ---- register prefetch of chunk 0 ----
    float pr[PPT][3];
#pragma unroll
    for (int k = 0; k < PPT; ++k) {
        const float* s = yc + (size_t)(k * THREADS + tid) * 3;
        pr[k][0] = s[0]; pr[k][1] = s[1]; pr[k][2] = s[2];
    }

    for (int c = 0; c < NCHUNK; ++c) {
        float4* buf = lbuf[c & 1];

        // transform + stage prefetched chunk into LDS
#pragma unroll
        for (int k = 0; k < PPT; ++k) {
            float y0 = pr[k][0], y1 = pr[k][1], y2 = pr[k][2];
            float4 p;
            p.x = -2.0f * y0;
            p.y = -2.0f * y1;
            p.z = -2.0f * y2;
            p.w = y0 * y0 + y1 * y1 + y2 * y2;
            buf[k * THREADS + tid] = p;                 // ds_store_b128
        }
        __syncthreads();   // single barrier per chunk is sufficient (double buffer)

        // kick off global loads for next chunk; latency hidden by compute below
        if (c + 1 < NCHUNK) {
            const float* base = yc + (size_t)(c + 1) * CH * 3;
#pragma unroll
            for (int k = 0; k < PPT; ++k) {
                const float* s = base + (size_t)(k * THREADS + tid) * 3;
                pr[k][0] = s[0]; pr[k][1] = s[1]; pr[k][2] = s[2];
            }
        }

        const int itbase = c * IPC;
#pragma unroll 2
        for (int jj = 0; jj < IPC; ++jj) {
            // candidate n = jj*16 + nlane of this chunk; b128 broadcast-friendly
            float4 p = buf[(jj << 4) + nlane];          // ds_load_b128
            v2f bv;
            bv[0] = lo ? p.x : p.z;                     // B: K=0 / K=2
            bv[1] = lo ? p.y : 0.0f;                    // B: K=1 / K=3 (pad)
            const int it = itbase + jj;                 // uniform iteration id
#pragma unroll
            for (int t = 0; t < TPW; ++t) {
                v8f cz = {0.f,0.f,0.f,0.f,0.f,0.f,0.f,0.f};
                // D = A x B : d'[m][n] = -2 * x_m . y_n
                v8f s = __builtin_amdgcn_wmma_f32_16x16x4_f32(
                    false, av[t], false, bv, (short)0, cz, false, false);
#pragma unroll
                for (int r = 0; r < 8; ++r) {
                    float d  = s[r] + p.w;              // + ||y_n||^2
                    bool  m_ = d < bd[t][r];
                    bi[t][r] = m_ ? it : bi[t][r];
                    bd[t][r] = m_ ? d  : bd[t][r];
                }
            }
        }
    }

    // ---- reduce across the 16 column lanes of each half-wave, write indices ----
#pragma unroll
    for (int t = 0; t < TPW; ++t) {
        float dv[8]; int ix[8];
#pragma unroll
        for (int r = 0; r < 8; ++r) {
            dv[r] = bd[t][r];
            ix[r] = (bi[t][r] << 4) + nlane;            // candidate index
        }
#pragma unroll
        for (int m = 1; m < 16; m <<= 1) {
#pragma unroll
            for (int r = 0; r < 8; ++r) {
                float od = __shfl_xor(dv[r], m, 16);
                int   oi = __shfl_xor(ix[r], m, 16);
                bool take = (od < dv[r]) || ((od == dv[r]) && (oi < ix[r]));
                dv[r] = take ? od : dv[r];
                ix[r] = take ? oi : ix[r];
            }
        }
        if (nlane == 0) {                // lane 0 -> rows 0-7, lane 16 -> rows 8-15
            int* o = outp + qb[t] + (lo ? 0 : 8);
#pragma unroll
            for (int r = 0; r < 8; ++r) o[r] = ix[r];
        }
    }
}

extern "C" void kernel_launch(void* const* d_in, const int* in_sizes, int n_in,
                              void* d_out, int out_size, void* d_ws, size_t ws_size,
                              hipStream_t stream) {
    (void)in_sizes; (void)n_in; (void)out_size; (void)d_ws; (void)ws_size;
    const float* xyz1 = (const float*)d_in[0];
    const float* xyz2 = (const float*)d_in[1];
    int* out = (int*)d_out;

    dim3 grid(2 * NB * WG_PER_PAIR);   // 1024 workgroups
    dim3 block(THREADS);               // 256 threads = 8 waves (wave32)
    chamfer_nn<<<grid, block, 0, stream>>>(xyz1, xyz2, out);
}